// BoundaryLoss_42322607735076
// MI455X (gfx1250) — compile-verified
//
#include <hip/hip_runtime.h>

#define HH 360
#define WW 640
#define BB 16
#define NC 7
#define TILE_W 64
#define TILE_H 4
#define HALO 2
#define LW (TILE_W + 2 * HALO) /* 68 */
#define LH (TILE_H + 2 * HALO) /* 8  */
#define HW (HH * WW)

__device__ __constant__ float kInvN = 1.0f / (float)(BB * HH * WW);

typedef float v2f __attribute__((ext_vector_type(2)));
typedef float v8f __attribute__((ext_vector_type(8)));

typedef __attribute__((address_space(1))) int* gptr_i32;
typedef __attribute__((address_space(3))) int* lptr_i32;

#if defined(__has_builtin)
#if __has_builtin(__builtin_amdgcn_global_load_async_to_lds_b32)
#define HAVE_ASYNC_LDS 1
#endif
#if __has_builtin(__builtin_amdgcn_wmma_f32_16x16x4_f32)
#define HAVE_WMMA_F32X4 1
#endif
#endif

__device__ __forceinline__ void wait_async_zero() {
#if defined(__has_builtin) && __has_builtin(__builtin_amdgcn_s_wait_asynccnt)
  __builtin_amdgcn_s_wait_asynccnt(0);
#else
  asm volatile("s_wait_asynccnt 0" ::: "memory");
#endif
}

// Fused kernel: stage target tile (with halo) into LDS via async-to-LDS,
// 17-tap ellipse max/min, 7-class log-softmax CE, weighted, block-reduced.
template <bool ATOMIC>
__global__ __launch_bounds__(256) void boundary_ce_kernel(
    const float* __restrict__ pred, const int* __restrict__ target,
    float* __restrict__ sink /* partial array, or out[0] if ATOMIC */) {
  __shared__ int tile[LH * LW];
  __shared__ float red[8];

  const int tid = threadIdx.x;
  const int x0 = blockIdx.x * TILE_W;
  const int y0 = blockIdx.y * TILE_H;
  const int b = blockIdx.z;
  const int* tb = target + (size_t)b * HW;

  // ---- stage halo tile into LDS (clamp-to-edge == ignore-OOB for this mask)
  for (int i = tid; i < LH * LW; i += 256) {
    int ty = i / LW, tx = i - ty * LW;
    int gy = min(max(y0 - HALO + ty, 0), HH - 1);
    int gx = min(max(x0 - HALO + tx, 0), WW - 1);
    const int* gp = tb + gy * WW + gx;
#ifdef HAVE_ASYNC_LDS
    __builtin_amdgcn_global_load_async_to_lds_b32(
        (gptr_i32)(int*)gp, (lptr_i32)&tile[i], 0, 0);
#else
    tile[i] = *gp;
#endif
  }
#ifdef HAVE_ASYNC_LDS
  wait_async_zero();
#endif
  __syncthreads();

  // ---- morphology (5x5 ellipse, 17 taps) from LDS
  const int lx = tid & 63;
  const int ly = tid >> 6;
  const int x = x0 + lx;
  const int y = y0 + ly;
  const int cx = lx + HALO;
  const int cy = ly + HALO;

  int mx = -0x7fffffff, mn = 0x7fffffff;
  {
    int v;
    v = tile[(cy - 2) * LW + cx]; mx = max(mx, v); mn = min(mn, v);
    v = tile[(cy + 2) * LW + cx]; mx = max(mx, v); mn = min(mn, v);
#pragma unroll
    for (int dy = -1; dy <= 1; ++dy) {
#pragma unroll
      for (int dx = -2; dx <= 2; ++dx) {
        v = tile[(cy + dy) * LW + cx + dx];
        mx = max(mx, v); mn = min(mn, v);
      }
    }
  }
  const float wgt = (mx != mn) ? 10.0f : 1.0f;
  const int t = tile[cy * LW + cx];

  // ---- log-softmax CE: pred streamed once, non-temporal
  const float* base = pred + (size_t)b * NC * HW + (size_t)y * WW + x;
  float p[NC];
#pragma unroll
  for (int c = 0; c < NC; ++c)
    p[c] = __builtin_nontemporal_load(base + (size_t)c * HW);
  float m = p[0];
#pragma unroll
  for (int c = 1; c < NC; ++c) m = fmaxf(m, p[c]);
  float s = 0.0f;
#pragma unroll
  for (int c = 0; c < NC; ++c) s += __expf(p[c] - m);
  const float ce = __logf(s) + m - p[t];
  float val = ce * wgt * kInvN;

  // ---- block reduction: wave32 shuffle + LDS across 8 waves
#pragma unroll
  for (int off = 16; off > 0; off >>= 1) val += __shfl_down(val, off, 32);
  if ((tid & 31) == 0) red[tid >> 5] = val;
  __syncthreads();
  if (tid == 0) {
    float bs = 0.0f;
#pragma unroll
    for (int i = 0; i < 8; ++i) bs += red[i];
    if (ATOMIC) {
      atomicAdd(sink, bs);
    } else {
      sink[((size_t)blockIdx.z * gridDim.y + blockIdx.y) * gridDim.x +
           blockIdx.x] = bs;
    }
  }
}

// Final deterministic reduction of per-block partials.
// Wave 0 reduces 1024 LDS floats with 16 chained V_WMMA_F32_16X16X4_F32
// (B = all-ones => D[m][n] = sum_k A[m][k] + C[m][n], f32 accumulate).
__global__ __launch_bounds__(1024) void final_reduce_kernel(
    const float* __restrict__ partial, int n, float* __restrict__ out) {
  __shared__ float buf[1024];
  const int tid = threadIdx.x;
  float s = 0.0f;
  for (int i = tid; i < n; i += 1024) s += partial[i];  // fixed order
  buf[tid] = s;
  __syncthreads();

  if (tid < 32) {  // wave 0 fully active -> EXEC all 1s for WMMA
#ifdef HAVE_WMMA_F32X4
    v8f acc = {};
    const v2f bones = {1.0f, 1.0f};
#pragma unroll
    for (int ch = 0; ch < 16; ++ch) {
      v2f a;
      a.x = buf[ch * 64 + 2 * tid];
      a.y = buf[ch * 64 + 2 * tid + 1];
      acc = __builtin_amdgcn_wmma_f32_16x16x4_f32(
          false, a, false, bones, (short)0, acc, false, false);
    }
    // every column of C holds the same row sums:
    // lanes 0-15 carry rows 0-7, lanes 16-31 carry rows 8-15.
    float t = acc[0] + acc[1] + acc[2] + acc[3] + acc[4] + acc[5] + acc[6] +
              acc[7];
    t += __shfl_xor(t, 16, 32);
    if (tid == 0) out[0] = t;
#else
    if (tid == 0) {
      float t = 0.0f;
      for (int i = 0; i < 1024; ++i) t += buf[i];
      out[0] = t;
    }
#endif
  }
}

__global__ void zero_out_kernel(float* out) { out[0] = 0.0f; }

extern "C" void kernel_launch(void* const* d_in, const int* in_sizes, int n_in,
                              void* d_out, int out_size, void* d_ws,
                              size_t ws_size, hipStream_t stream) {
  (void)in_sizes; (void)n_in; (void)out_size;
  const float* pred = (const float*)d_in[0];
  const int* target = (const int*)d_in[1];
  float* out = (float*)d_out;

  dim3 grid(WW / TILE_W, HH / TILE_H, BB);  // 10 x 90 x 16 = 14400 blocks
  const size_t nblocks = (size_t)grid.x * grid.y * grid.z;

  if (ws_size >= nblocks * sizeof(float)) {
    float* partial = (float*)d_ws;
    boundary_ce_kernel<false><<<grid, 256, 0, stream>>>(pred, target, partial);
    final_reduce_kernel<<<1, 1024, 0, stream>>>(partial, (int)nblocks, out);
  } else {
    zero_out_kernel<<<1, 1, 0, stream>>>(out);
    boundary_ce_kernel<true><<<grid, 256, 0, stream>>>(pred, target, out);
  }
}